// Jeffrey_13984413515966
// MI455X (gfx1250) — compile-verified
//
#include <hip/hip_runtime.h>
#include <hip/hip_bf16.h>

// ---------------------------------------------------------------------------
// Pairwise-distance Jeffrey loss, algebraically collapsed to one streaming
// pass over X (8192x512 f32) + O(1) finale. MI455X (gfx1250, wave32).
//
//   S_i = n*sq_i + SQ - 2<x_i,m> + 2eps(n*s_i - Ssum) + n*d*eps^2
//   sum_i c_i S_i needs only  <p,m>, sum c_i sq_i, sum c_i s_i  with
//   m = sum_i x_i,  p = sum_i c_i x_i,  c_i = hist[target_i].
//
// ws layout (float words):
//   [0,512)    m        column sums
//   [512,1024) p        c-weighted column sums
//   [1024]     SQ   = sum x^2          [1025] CSQ = sum c_i * x^2
//   int words at float offset 1028: hist[128]
// ---------------------------------------------------------------------------

typedef __attribute__((ext_vector_type(2))) float v2f;
typedef __attribute__((ext_vector_type(8))) float v8f;

#define N_ROWS 8192
#define N_COLS 512
#define JEPS   1e-6f
#define NBINS  128          // 100 label bins used, padded to power of two
#define ROWS_PER_BLOCK 32
#define WS_WORDS (1028 + NBINS)

// ---------------- kernel 0: zero the workspace ------------------------------
__global__ void k_zero(int* __restrict__ w) {
    int i = blockIdx.x * blockDim.x + threadIdx.x;
    if (i < WS_WORDS) w[i] = 0;   // 0x0 == 0.0f for the float region too
}

// ---------------- kernel 1: label histogram ---------------------------------
__global__ void k_hist(const int* __restrict__ targets, int* __restrict__ hist) {
    __shared__ int lh[NBINS];
    for (int i = threadIdx.x; i < NBINS; i += blockDim.x) lh[i] = 0;
    __syncthreads();
    int idx = blockIdx.x * blockDim.x + threadIdx.x;
    if (idx < N_ROWS) atomicAdd(&lh[targets[idx] & (NBINS - 1)], 1);
    __syncthreads();
    for (int i = threadIdx.x; i < NBINS; i += blockDim.x)
        if (lh[i]) atomicAdd(&hist[i], lh[i]);
}

// ---------------- kernel 2: single streaming pass over X --------------------
// blockDim = 128 (4 waves); thread t owns columns 4t..4t+3 (float4 loads);
// each block consumes ROWS_PER_BLOCK consecutive rows (fully coalesced b128).
__global__ void k_stream(const float4* __restrict__ X4,
                         const int* __restrict__ targets,
                         const int* __restrict__ hist,
                         float* __restrict__ m, float* __restrict__ p,
                         float* __restrict__ scal) {
    const int t    = threadIdx.x;                  // 0..127
    const int row0 = blockIdx.x * ROWS_PER_BLOCK;

    float4 msum = make_float4(0.f, 0.f, 0.f, 0.f);
    float4 psum = make_float4(0.f, 0.f, 0.f, 0.f);
    float  sq = 0.f, csq = 0.f;

    #pragma unroll 4
    for (int r = 0; r < ROWS_PER_BLOCK; ++r) {
        const int   row = row0 + r;
        const float c   = (float)hist[targets[row] & (NBINS - 1)];  // uniform
        const float4 x  = X4[(size_t)row * (N_COLS / 4) + t];
        msum.x += x.x;      msum.y += x.y;      msum.z += x.z;      msum.w += x.w;
        psum.x += c * x.x;  psum.y += c * x.y;  psum.z += c * x.z;  psum.w += c * x.w;
        const float xs = x.x * x.x + x.y * x.y + x.z * x.z + x.w * x.w;
        sq  += xs;
        csq += c * xs;
    }

    atomicAdd(&m[4 * t + 0], msum.x);  atomicAdd(&m[4 * t + 1], msum.y);
    atomicAdd(&m[4 * t + 2], msum.z);  atomicAdd(&m[4 * t + 3], msum.w);
    atomicAdd(&p[4 * t + 0], psum.x);  atomicAdd(&p[4 * t + 1], psum.y);
    atomicAdd(&p[4 * t + 2], psum.z);  atomicAdd(&p[4 * t + 3], psum.w);

    // wave32 shuffle reduce, then cross-wave via LDS, one atomic per block
    for (int off = 16; off > 0; off >>= 1) {
        sq  += __shfl_down(sq,  off);
        csq += __shfl_down(csq, off);
    }
    __shared__ float red[2][4];
    const int wave = t >> 5, lane = t & 31;
    if (lane == 0) { red[0][wave] = sq; red[1][wave] = csq; }
    __syncthreads();
    if (t == 0) {
        float a = 0.f, b = 0.f;
        #pragma unroll
        for (int w = 0; w < 4; ++w) { a += red[0][w]; b += red[1][w]; }
        atomicAdd(&scal[0], a);
        atomicAdd(&scal[1], b);
    }
}

// ---------------- finale: one wave, WMMA dot products + scalar assembly -----
// Diagonal of accumulated 16x16 f32 C tile (ISA 7.12.2):
//   (i,i) i<8  -> VGPR i,   lane i
//   (i,i) i>=8 -> VGPR i-8, lane i+16   (lanes 24..31)
__device__ __forceinline__ float diag_extract(v8f c, int lane) {
    const int  idx   = (lane < 8) ? lane : (lane - 24);
    const bool valid = (lane < 8) || (lane >= 24);
    float t = c[0];
    if (idx == 1) t = c[1];
    if (idx == 2) t = c[2];
    if (idx == 3) t = c[3];
    if (idx == 4) t = c[4];
    if (idx == 5) t = c[5];
    if (idx == 6) t = c[6];
    if (idx == 7) t = c[7];
    return valid ? t : 0.f;
}

__global__ void k_final(const float* __restrict__ ws,
                        const int* __restrict__ hist,
                        float* __restrict__ out) {
    const float* m = ws;
    const float* p = ws + 512;
    const float  SQ  = ws[1024];
    const float  CSQ = ws[1025];
    const int lane = threadIdx.x;            // exactly 32 lanes, EXEC all-ones

    // A[r][k] = u[64g + 4r + k], B[k][c] = v[64g + 4c + k]  ->  diag(D) gives
    // aligned 4-chunk dots; accumulate C over 8 chunks, read diagonal once.
    const int r  = lane & 15;
    const int kb = (lane < 16) ? 0 : 2;      // 32-bit A 16x4 lane-half K split
    v8f cmm = {};                            // <m,m> accumulator
    v8f cpm = {};                            // <p,m> accumulator
    #pragma unroll
    for (int g = 0; g < 8; ++g) {
        const int ai = 64 * g + 4 * r + kb;
        v2f am = { m[ai], m[ai + 1] };       // doubles as B operand (c == r)
        v2f ap = { p[ai], p[ai + 1] };
        cmm = __builtin_amdgcn_wmma_f32_16x16x4_f32(
                  false, am, false, am, (short)0, cmm, false, false);
        cpm = __builtin_amdgcn_wmma_f32_16x16x4_f32(
                  false, ap, false, am, (short)0, cpm, false, false);
    }
    float dmm = diag_extract(cmm, lane);
    float dpm = diag_extract(cpm, lane);

    // Ssum = sum m, CS = sum p, sumC = sum hist[b]^2
    float ssum = 0.f, cs = 0.f, sc = 0.f;
    #pragma unroll
    for (int i = lane; i < N_COLS; i += 32) { ssum += m[i]; cs += p[i]; }
    #pragma unroll
    for (int i = lane; i < NBINS; i += 32) { float h = (float)hist[i]; sc += h * h; }

    for (int off = 16; off > 0; off >>= 1) {
        dmm  += __shfl_down(dmm,  off);
        dpm  += __shfl_down(dpm,  off);
        ssum += __shfl_down(ssum, off);
        cs   += __shfl_down(cs,   off);
        sc   += __shfl_down(sc,   off);
    }

    if (lane == 0) {
        const float nf   = (float)N_ROWS;
        const float df   = (float)N_COLS;
        const float ne2  = nf * df * JEPS * JEPS;           // n*d*eps^2
        const float sumC = sc;
        const float sumD = nf * nf - sumC;
        // sum_i S_i  and  sum_i c_i S_i  in closed form
        const float sumS  = 2.f * nf * SQ - 2.f * dmm + nf * ne2;
        const float sumCS = nf * CSQ + SQ * sumC - 2.f * dpm
                          + 2.f * JEPS * (nf * cs - ssum * sumC) + ne2 * sumC;
        const float sumDS = nf * sumS - sumCS;
        const float a = sumCS / (nf * sumC);
        const float b = sumDS / (nf * sumD);
        out[0] = -a / b - b / a;
    }
}

// ---------------------------------------------------------------------------
extern "C" void kernel_launch(void* const* d_in, const int* in_sizes, int n_in,
                              void* d_out, int out_size, void* d_ws, size_t ws_size,
                              hipStream_t stream) {
    (void)in_sizes; (void)n_in; (void)out_size; (void)ws_size;
    const float* X  = (const float*)d_in[0];
    const int*   tg = (const int*)d_in[1];
    float* ws   = (float*)d_ws;
    float* m    = ws;
    float* p    = ws + 512;
    float* scal = ws + 1024;
    int*   hist = (int*)(ws + 1028);
    float* out  = (float*)d_out;

    k_zero  <<<(WS_WORDS + 255) / 256, 256, 0, stream>>>((int*)d_ws);
    k_hist  <<<N_ROWS / 256, 256, 0, stream>>>(tg, hist);
    k_stream<<<N_ROWS / ROWS_PER_BLOCK, 128, 0, stream>>>(
        (const float4*)X, tg, hist, m, p, scal);
    k_final <<<1, 32, 0, stream>>>(ws, hist, out);
}